// CoefficientMaxPool_33036888441089
// MI455X (gfx1250) — compile-verified
//
#include <hip/hip_runtime.h>
#include <stdint.h>

// CoefficientMaxPool for MI455X (gfx1250).
// Memory-bound argmax/gather: 128 MiB streamed -> ~5.5 us floor at 23.3 TB/s.
// Uses the CDNA5 async global->LDS DMA path (global_load_async_to_lds_b128,
// GVS addressing: SADDR sgpr64 + VADDR vgpr32 + IOFFSET imm applied to BOTH
// the global and LDS side) with double buffering and s_wait_asynccnt.

#define NSPLIT 16            // n-range splits per batch (512/16 = 32 n per block)
#define TN 4                 // neighbors per LDS tile
#define TILE_FLOATS (TN * 16 * 128)            // 8192 floats = 32 KB
#define TILE_BYTES (TILE_FLOATS * 4)
#define THREADS1 128
#define CHUNKS_PER_THREAD (TILE_BYTES / 16 / THREADS1)  // 16 async b128 / thread / tile
#define CHUNK_STRIDE (THREADS1 * 16)                    // 2048 B between a thread's chunks
#define NPART (32 * NSPLIT * 4 * 128)          // 262144 partial (max,idx) pairs

// Issue CHUNKS_PER_THREAD async b128 copies. GVS form:
//   chunk K copies 16 bytes from global address sBase + vOff + K*2048 to LDS address ldsOff + K*2048
// Template recursion guarantees the offset is a compile-time immediate.
template <int K>
__device__ __forceinline__ void issue_chunks(uint32_t ldsOff, uint32_t vOff,
                                             uint64_t sBase) {
    asm volatile("global_load_async_to_lds_b128 %0, %1, %2 offset:%3"
                 :: "v"(ldsOff), "v"(vOff), "s"(sBase), "i"(K * CHUNK_STRIDE)
                 : "memory");
    if constexpr (K + 1 < CHUNKS_PER_THREAD)
        issue_chunks<K + 1>(ldsOff, vOff, sBase);
}

__global__ __launch_bounds__(THREADS1)
void cmp_phase1(const float* __restrict__ x,
                float* __restrict__ wmax, int* __restrict__ widx) {
    __shared__ __align__(16) float smem[2 * TILE_FLOATS];   // 64 KB double buffer

    const int ns  = blockIdx.x;        // 0..15
    const int b   = blockIdx.y;        // 0..31
    const int tid = threadIdx.x;
    const int cx  = tid & 31;          // channel quad 0..31 (c = cx*4 .. cx*4+3)
    const int ly  = tid >> 5;          // n-lane 0..3
    const int n0  = ns * (512 / NSPLIT);
    const int numTiles = (512 / NSPLIT) / TN;   // 8

    const uint64_t xBase = (uint64_t)(uintptr_t)x;
    // Flat LDS addresses carry the byte offset in the low 32 bits.
    const uint32_t ldsBase = (uint32_t)(uintptr_t)(&smem[0]);
    const uint32_t vOff    = (uint32_t)tid * 16u;          // per-lane chunk offset
    const uint32_t ldsOff0 = ldsBase + vOff;               // buffer 0
    const uint32_t ldsOff1 = ldsBase + (uint32_t)TILE_BYTES + vOff;  // buffer 1

    constexpr int L_OF_I[16] = {0, 1,1,1, 2,2,2,2,2, 3,3,3,3,3,3,3};

    auto issue_tile = [&](int t) {
        // Tile of TN consecutive neighbors is a contiguous 32 KB region.
        const uint64_t gTile = xBase +
            ((uint64_t)((b * 512 + n0 + t * TN)) * 2048ull) * 4ull;  // uniform -> SGPR
        issue_chunks<0>((t & 1) ? ldsOff1 : ldsOff0, vOff, gTile);
    };

    float best[4][4];
    int   bidx[4][4];
#pragma unroll
    for (int l = 0; l < 4; ++l)
#pragma unroll
        for (int j = 0; j < 4; ++j) { best[l][j] = -1.0f; bidx[l][j] = 0; }

    issue_tile(0);

    for (int t = 0; t < numTiles; ++t) {
        if (t + 1 < numTiles) {
            issue_tile(t + 1);
            // Async loads complete in order: <=16 outstanding => tile t landed.
            asm volatile("s_wait_asynccnt 16" ::: "memory");
        } else {
            asm volatile("s_wait_asynccnt 0" ::: "memory");
        }
        __syncthreads();   // tile t visible to all waves

        const float4* buf =
            reinterpret_cast<const float4*>(smem + (t & 1) * TILE_FLOATS);
        const int n = n0 + t * TN + ly;   // ascending n => strict '>' keeps first max

        float ls[4][4];
#pragma unroll
        for (int l = 0; l < 4; ++l)
            { ls[l][0] = 0.f; ls[l][1] = 0.f; ls[l][2] = 0.f; ls[l][3] = 0.f; }
#pragma unroll
        for (int i = 0; i < 16; ++i) {
            float4 v = buf[(ly * 16 + i) * 32 + cx];
            const int l = L_OF_I[i];
            ls[l][0] += v.x * v.x;
            ls[l][1] += v.y * v.y;
            ls[l][2] += v.z * v.z;
            ls[l][3] += v.w * v.w;
        }
#pragma unroll
        for (int l = 0; l < 4; ++l)
#pragma unroll
            for (int j = 0; j < 4; ++j)
                if (ls[l][j] > best[l][j]) { best[l][j] = ls[l][j]; bidx[l][j] = n; }

        __syncthreads();   // all reads of buf[t&1] done before it is re-filled
    }

    // Block reduction across the 4 n-lanes; reuse the (now idle) tile buffer.
    float* smax = smem;                          // [ly][l][c] : 2048 floats
    int*   sidx = (int*)(smem + 2048);           // [ly][l][c] : 2048 ints
#pragma unroll
    for (int l = 0; l < 4; ++l)
#pragma unroll
        for (int j = 0; j < 4; ++j) {
            const int c = cx * 4 + j;
            smax[(ly * 4 + l) * 128 + c] = best[l][j];
            sidx[(ly * 4 + l) * 128 + c] = bidx[l][j];
        }
    __syncthreads();

    for (int s = tid; s < 512; s += THREADS1) {
        const int l = s >> 7, c = s & 127;
        float bv = -1.0f; int bi = 0x7fffffff;
#pragma unroll
        for (int y = 0; y < 4; ++y) {
            const float v  = smax[(y * 4 + l) * 128 + c];
            const int   iv = sidx[(y * 4 + l) * 128 + c];
            if (v > bv || (v == bv && iv < bi)) { bv = v; bi = iv; }
        }
        const int p = ((b * NSPLIT + ns) * 4 + l) * 128 + c;
        wmax[p] = bv;
        widx[p] = bi;
    }
}

__global__ __launch_bounds__(512)
void cmp_phase2(const float* __restrict__ x,
                const float* __restrict__ wmax, const int* __restrict__ widx,
                float* __restrict__ out) {
    __shared__ int sIdx[512];   // [l][c]
    const int b = blockIdx.x, tid = threadIdx.x;

    {   // reduce the NSPLIT partials for this thread's (l, c)
        const int l = tid >> 7, c = tid & 127;
        float bv = -1.0f; int bi = 0x7fffffff;
        for (int s = 0; s < NSPLIT; ++s) {
            const int p = ((b * NSPLIT + s) * 4 + l) * 128 + c;
            const float v  = wmax[p];
            const int   iv = widx[p];
            if (v > bv || (v == bv && iv < bi)) { bv = v; bi = iv; }
        }
        sIdx[tid] = bi;
    }
    __syncthreads();

    // Gather the winning neighbor per (irrep, channel): 2048 outputs / block.
#pragma unroll
    for (int r = 0; r < 4; ++r) {
        const int o = r * 512 + tid;
        const int i = o >> 7, c = o & 127;
        const int l = (i >= 9) ? 3 : (i >= 4) ? 2 : (i >= 1) ? 1 : 0;
        const int n = sIdx[l * 128 + c];
        out[(b * 16 + i) * 128 + c] =
            x[(((size_t)(b * 512 + n)) * 16 + (size_t)i) * 128 + (size_t)c];
    }
}

extern "C" void kernel_launch(void* const* d_in, const int* in_sizes, int n_in,
                              void* d_out, int out_size, void* d_ws, size_t ws_size,
                              hipStream_t stream) {
    const float* x = (const float*)d_in[0];
    // d_in[1] (i2l, 16x4 indicator) is a fixed LMAX=3 mapping; hardcoded above.
    float* wmax = (float*)d_ws;                       // NPART floats  (1 MB)
    int*   widx = (int*)((float*)d_ws + NPART);       // NPART ints    (1 MB)
    float* out  = (float*)d_out;                      // [32,16,128]

    cmp_phase1<<<dim3(NSPLIT, 32), THREADS1, 0, stream>>>(x, wmax, widx);
    cmp_phase2<<<32, 512, 0, stream>>>(x, wmax, widx, out);
}